// HypergraphAttentionNetwork_77077483094551
// MI455X (gfx1250) — compile-verified
//
#include <hip/hip_runtime.h>
#include <hip/hip_bf16.h>

#define N_NODES  200000
#define N_EDGES  20000
#define NNZ      2000000
#define IN_CH    128
#define HID      32
#define AOUT     64

typedef __attribute__((ext_vector_type(2))) float v2f;
typedef __attribute__((ext_vector_type(8))) float v8f;

// D = A(16x4 f32) * B(4x16 f32) + C(16x16 f32)
// args: (neg_a, A, neg_b, B, c_mod, C, reuse_a, reuse_b)
__device__ __forceinline__ v8f wmma4(v2f a, v2f b, v8f c) {
    return __builtin_amdgcn_wmma_f32_16x16x4_f32(false, a, false, b, (short)0, c,
                                                 false, false);
}

// ---------------------------------------------------------------- zero fill
__global__ __launch_bounds__(256) void k_zero(float* __restrict__ p, size_t n) {
    size_t i = (size_t)blockIdx.x * blockDim.x + threadIdx.x;
    size_t stride = (size_t)gridDim.x * blockDim.x;
    for (; i < n; i += stride) p[i] = 0.0f;
}

// ---------------------------------------------------------------- xw = x @ Wconv
// One wave computes a 16-row x 32-col tile: two 16x16 accumulators, K=128 in
// steps of 4 with V_WMMA_F32_16X16X4_F32. Wconv (128x32, 16KB) staged in LDS.
__global__ __launch_bounds__(256) void k_gemm_xw(const float* __restrict__ x,
                                                 const float* __restrict__ w,
                                                 float* __restrict__ xw) {
    __shared__ float ldsW[IN_CH * HID];  // 16 KB
    for (int i = threadIdx.x; i < IN_CH * HID; i += 256) ldsW[i] = w[i];
    __syncthreads();

    const int wave = threadIdx.x >> 5;
    const int lane = threadIdx.x & 31;
    const int rowBase = (blockIdx.x * 8 + wave) * 16;
    if (rowBase >= N_NODES) return;

    const int row  = lane & 15;        // A: M index
    const int kh   = (lane >> 4) << 1; // A/B: +0 or +2 in K within each 4-chunk
    const int col  = lane & 15;        // B: N index
    const int half = lane >> 4;        // C: M-half select

    const float* xr = x + (size_t)(rowBase + row) * IN_CH;
    __builtin_prefetch(xr, 0, 3);            // global_prefetch_b8
    __builtin_prefetch(xr + 64, 0, 3);

    v8f acc0 = {}; v8f acc1 = {};
#pragma unroll
    for (int k = 0; k < IN_CH; k += 4) {
        const int kk = k + kh;
        v2f a;  a.x  = xr[kk];                    a.y  = xr[kk + 1];
        v2f b0; b0.x = ldsW[kk * HID + col];      b0.y = ldsW[(kk + 1) * HID + col];
        v2f b1; b1.x = ldsW[kk * HID + col + 16]; b1.y = ldsW[(kk + 1) * HID + col + 16];
        acc0 = wmma4(a, b0, acc0);
        acc1 = wmma4(a, b1, acc1);
    }

    float* ob = xw + (size_t)rowBase * HID;
#pragma unroll
    for (int i = 0; i < 8; ++i) {
        const int m = i + half * 8;
        ob[(size_t)m * HID + col]      = acc0[i];
        ob[(size_t)m * HID + col + 16] = acc1[i];
    }
}

// ---------------------------------------------------------------- scatter-add
// Wave per nnz, lane = channel (HID==32 == wave32). nnz id is forced into an
// SGPR via readfirstlane so the two int64 index fetches become broadcast
// s_load_b64 (SMEM) instead of 32-lane vector loads. Row gather and the 32
// global_atomic_add_f32 are fully coalesced and L2-resident. Lane 0 also
// accumulates the destination-degree count (+1 per nnz), replacing the
// standalone degree kernel.
__global__ __launch_bounds__(256) void k_scatter(const float* __restrict__ src,
                                                 const long long* __restrict__ srcIdx,
                                                 const long long* __restrict__ dstIdx,
                                                 float* __restrict__ dst,
                                                 float* __restrict__ dstDeg) {
    const int i = __builtin_amdgcn_readfirstlane(blockIdx.x * 8 + (threadIdx.x >> 5));
    const int c = threadIdx.x & 31;
    if (i < NNZ) {
        const int s = (int)srcIdx[i];
        const int d = (int)dstIdx[i];
        unsafeAtomicAdd(&dst[(size_t)d * HID + c], src[(size_t)s * HID + c]);
        if (c == 0) unsafeAtomicAdd(&dstDeg[d], 1.0f);
    }
}

// ---------------------------------------------------------------- e *= Binv
__global__ __launch_bounds__(256) void k_scale_e(float* __restrict__ e,
                                                 const float* __restrict__ B) {
    int gid = blockIdx.x * 256 + threadIdx.x;
    if (gid < N_EDGES * HID) {
        const float b = B[gid >> 5];
        e[gid] *= (b > 0.0f) ? (1.0f / b) : 0.0f;
    }
}

// ---------------------------------------------------------------- h = relu(Dinv*acc + bias)
__global__ __launch_bounds__(256) void k_finalize_h(const float* __restrict__ hacc,
                                                    const float* __restrict__ D,
                                                    const float* __restrict__ bias,
                                                    float* __restrict__ h) {
    long long gid = (long long)blockIdx.x * 256 + threadIdx.x;
    if (gid < (long long)N_NODES * HID) {
        const float d = D[gid >> 5];
        const float inv = (d > 0.0f) ? (1.0f / d) : 0.0f;
        float v = hacc[gid] * inv + bias[gid & 31];
        h[gid] = v > 0.0f ? v : 0.0f;
    }
}

// ---------------------------------------------------------------- attention + fc
// seq len == 1  =>  softmax over singleton axis == 1  =>  attention is identity.
// out[n] = sum_j relu( (h @ Wattn)[n,j] + battn[j] ) * fcw[j] + fcb
// One wave: 16 nodes, 4 WMMA column tiles (N=64), K=32 in 8 steps of f32 WMMA.
__global__ __launch_bounds__(256) void k_attn_fc(const float* __restrict__ h,
                                                 const float* __restrict__ attnW,
                                                 const float* __restrict__ attnB,
                                                 const float* __restrict__ fcW,
                                                 const float* __restrict__ fcB,
                                                 float* __restrict__ out) {
    __shared__ float ldsW[HID * AOUT];  // 8 KB
    for (int i = threadIdx.x; i < HID * AOUT; i += 256) ldsW[i] = attnW[i];
    __syncthreads();

    const int wave = threadIdx.x >> 5;
    const int lane = threadIdx.x & 31;
    const int rowBase = (blockIdx.x * 8 + wave) * 16;
    if (rowBase >= N_NODES) return;

    const int row  = lane & 15;
    const int kh   = (lane >> 4) << 1;
    const int col  = lane & 15;
    const int half = lane >> 4;

    const float* hr = h + (size_t)(rowBase + row) * HID;
    __builtin_prefetch(hr, 0, 3);

    v8f acc[4] = {};
#pragma unroll
    for (int k = 0; k < HID; k += 4) {
        const int kk = k + kh;
        v2f a; a.x = hr[kk]; a.y = hr[kk + 1];
#pragma unroll
        for (int g = 0; g < 4; ++g) {
            v2f b;
            b.x = ldsW[kk * AOUT + col + g * 16];
            b.y = ldsW[(kk + 1) * AOUT + col + g * 16];
            acc[g] = wmma4(a, b, acc[g]);
        }
    }

    float bs[4], fw[4];
#pragma unroll
    for (int g = 0; g < 4; ++g) {
        const int n = col + g * 16;
        bs[g] = attnB[n];
        fw[g] = fcW[n];
    }
    const float fb = fcB[0];

#pragma unroll
    for (int i = 0; i < 8; ++i) {
        float s = 0.0f;
#pragma unroll
        for (int g = 0; g < 4; ++g) {
            float v = acc[g][i] + bs[g];
            v = v > 0.0f ? v : 0.0f;
            s += v * fw[g];
        }
        // reduce across the 16 lanes of this C-matrix half (N dimension)
        for (int m = 1; m < 16; m <<= 1) s += __shfl_xor(s, m, 32);
        if (col == 0) out[rowBase + i + half * 8] = s + fb;
    }
}

// ---------------------------------------------------------------- launch
extern "C" void kernel_launch(void* const* d_in, const int* in_sizes, int n_in,
                              void* d_out, int out_size, void* d_ws, size_t ws_size,
                              hipStream_t stream) {
    const float*     x     = (const float*)d_in[0];
    const long long* hidx  = (const long long*)d_in[1];  // int64, shape (2, NNZ)
    const long long* nidx  = hidx;
    const long long* eidx  = hidx + NNZ;
    const float*     wconv = (const float*)d_in[2];
    const float*     bconv = (const float*)d_in[3];
    const float*     attnW = (const float*)d_in[4];
    const float*     attnB = (const float*)d_in[5];
    // d_in[6..9] (a1_w,a1_b,a2_w,a2_b) are dead: softmax over a length-1 axis == 1
    const float*     fcW   = (const float*)d_in[10];
    const float*     fcB   = (const float*)d_in[11];
    float*           out   = (float*)d_out;

    // Workspace layout (floats). Accumulators first so one zero pass covers them.
    float* ws   = (float*)d_ws;
    float* D    = ws;                                   // N_NODES   (node degree)
    float* B    = D    + N_NODES;                       // N_EDGES   (edge degree)
    float* e    = B    + N_EDGES;                       // N_EDGES*HID
    float* hacc = e    + (size_t)N_EDGES * HID;         // N_NODES*HID
    float* xw   = hacc + (size_t)N_NODES * HID;         // N_NODES*HID
    float* h    = xw   + (size_t)N_NODES * HID;         // N_NODES*HID
    const size_t zeroSpan = (size_t)N_NODES + N_EDGES +
                            (size_t)N_EDGES * HID + (size_t)N_NODES * HID;

    k_zero<<<4096, 256, 0, stream>>>(ws, zeroSpan);

    const int gemmBlocks = (N_NODES / 16 + 7) / 8;  // 8 waves/block, 16 rows/wave
    k_gemm_xw<<<gemmBlocks, 256, 0, stream>>>(x, wconv, xw);

    const int scatBlocks = (NNZ + 7) / 8;           // wave per nnz, 8 waves/block
    // pass 1: nodes -> edges; piggy-back edge-degree B
    k_scatter<<<scatBlocks, 256, 0, stream>>>(xw, nidx, eidx, e, B);

    k_scale_e<<<(N_EDGES * HID + 255) / 256, 256, 0, stream>>>(e, B);

    // pass 2: edges -> nodes; piggy-back node-degree D
    k_scatter<<<scatBlocks, 256, 0, stream>>>(e, eidx, nidx, hacc, D);

    const long long finThreads = (long long)N_NODES * HID;
    k_finalize_h<<<(int)((finThreads + 255) / 256), 256, 0, stream>>>(hacc, D, bconv, h);

    k_attn_fc<<<gemmBlocks, 256, 0, stream>>>(h, attnW, attnB, fcW, fcB, out);
}